// MixEHR_Seed_56461640073240
// MI455X (gfx1250) — compile-verified
//
#include <hip/hip_runtime.h>
#include <hip/hip_bf16.h>
#include <math.h>

#define ETA    0.1f
#define BETA   0.1f
#define MU     0.1f
#define PI_INIT 0.7f
#define MINI   1e-6f
#define C_TOTAL 10000000.0f

#define MAXB 64
#define MAXK 64
#define VPB  32     // v's per block in the main kernel
#define BOWP 33     // padded stride for bow tile (bank-conflict free)

typedef __attribute__((ext_vector_type(2))) float v2f;
typedef __attribute__((ext_vector_type(8))) float v8f;

// ---------------------------------------------------------------------------
// 0) zero workspace accumulators
// ---------------------------------------------------------------------------
__global__ void zero_kernel(float* p, int n) {
    for (int i = blockIdx.x * blockDim.x + threadIdx.x; i < n;
         i += gridDim.x * blockDim.x)
        p[i] = 0.0f;
}

// ---------------------------------------------------------------------------
// 1) seed_word[v] = (sum_k seeds[v,k] > 0); S = count(seed_word)
// ---------------------------------------------------------------------------
__global__ void seed_word_kernel(const float* __restrict__ seeds,
                                 float* __restrict__ seed_word,
                                 int* __restrict__ S, int V, int K) {
    int v = blockIdx.x * blockDim.x + threadIdx.x;
    int cnt = 0;
    if (v < V) {
        float s = 0.0f;
        for (int k = 0; k < K; ++k) s += seeds[v * K + k];
        float sw = (s > 0.0f) ? 1.0f : 0.0f;
        seed_word[v] = sw;
        cnt = (int)sw;
    }
    __shared__ int red[256];
    red[threadIdx.x] = cnt;
    __syncthreads();
    for (int s = blockDim.x / 2; s > 0; s >>= 1) {
        if (threadIdx.x < s) red[threadIdx.x] += red[threadIdx.x + s];
        __syncthreads();
    }
    if (threadIdx.x == 0) atomicAdd(S, red[0]);
}

// ---------------------------------------------------------------------------
// 2) batch_C = sum(bow)  (exact integer sum)
// ---------------------------------------------------------------------------
__global__ void batchc_kernel(const int* __restrict__ bow, int* __restrict__ bc,
                              int n) {
    int acc = 0;
    for (int i = blockIdx.x * blockDim.x + threadIdx.x; i < n;
         i += gridDim.x * blockDim.x)
        acc += bow[i];
    __shared__ int red[256];
    red[threadIdx.x] = acc;
    __syncthreads();
    for (int s = blockDim.x / 2; s > 0; s >>= 1) {
        if (threadIdx.x < s) red[threadIdx.x] += red[threadIdx.x + s];
        __syncthreads();
    }
    if (threadIdx.x == 0) atomicAdd(bc, red[0]);
}

// ---------------------------------------------------------------------------
// 3) column sums via V_WMMA_F32_16X16X4_F32 ones-trick:
//    A(16x4)=ones  =>  D[m,n] = sum_j B[j,n]  (k-slot permutation invariant).
//    Each wave owns one 16-wide k tile; blocks stride over v in chunks of 8
//    (2 WMMAs / iter, unconditional loads in the steady-state loop).
//    out[] must be pre-zeroed (accumulated with global atomics).
// ---------------------------------------------------------------------------
__global__ void colsum_wmma_kernel(const float* __restrict__ M,
                                   float* __restrict__ out, int V, int K) {
    int t    = threadIdx.x;
    int wave = t >> 5;          // 0..3 -> k tile
    int lane = t & 31;
    int n    = lane & 15;
    int half = lane >> 4;
    int kk   = wave * 16 + n;
    float kmask = (kk < K) ? 1.0f : 0.0f;
    int   kidx  = (kk < K) ? kk : (K - 1);

    v2f a; a.x = 1.0f; a.y = 1.0f;          // ones A-matrix
    v8f c = {0, 0, 0, 0, 0, 0, 0, 0};

    const int stride = gridDim.x * 8;
    int v0 = blockIdx.x * 8;

    // steady state: whole 8-row chunk in bounds -> unconditional loads,
    // 4 independent loads feed 2 chained WMMAs.
    for (; v0 + 8 <= V; v0 += stride) {
        const float* p = M + v0 * K + kidx;
        v2f b0, b1;
        b0.x = p[(half)     * K] * kmask;
        b0.y = p[(half + 2) * K] * kmask;
        b1.x = p[(half + 4) * K] * kmask;
        b1.y = p[(half + 6) * K] * kmask;
        c = __builtin_amdgcn_wmma_f32_16x16x4_f32(false, a, false, b0,
                                                  (short)0, c, false, false);
        c = __builtin_amdgcn_wmma_f32_16x16x4_f32(false, a, false, b1,
                                                  (short)0, c, false, false);
    }
    // tail: at most one partial chunk per block, branchless clamp+mask loads
    if (v0 < V) {
        int r0 = v0 + half, r1 = v0 + 2 + half, r2 = v0 + 4 + half,
            r3 = v0 + 6 + half;
        float m0 = (r0 < V) ? kmask : 0.0f;
        float m1 = (r1 < V) ? kmask : 0.0f;
        float m2 = (r2 < V) ? kmask : 0.0f;
        float m3 = (r3 < V) ? kmask : 0.0f;
        int c0 = (r0 < V) ? r0 : (V - 1);
        int c1 = (r1 < V) ? r1 : (V - 1);
        int c2 = (r2 < V) ? r2 : (V - 1);
        int c3 = (r3 < V) ? r3 : (V - 1);
        v2f b0, b1;
        b0.x = M[c0 * K + kidx] * m0;
        b0.y = M[c1 * K + kidx] * m1;
        b1.x = M[c2 * K + kidx] * m2;
        b1.y = M[c3 * K + kidx] * m3;
        c = __builtin_amdgcn_wmma_f32_16x16x4_f32(false, a, false, b0,
                                                  (short)0, c, false, false);
        c = __builtin_amdgcn_wmma_f32_16x16x4_f32(false, a, false, b1,
                                                  (short)0, c, false, false);
    }
    // D rows are identical (A rows identical); lane<16 of VGPR0 holds N=lane.
    if (lane < 16 && kk < K) atomicAdd(&out[kk], c[0]);
}

// ---------------------------------------------------------------------------
// 4) main fused kernel: gamma computation + normalization + all reductions
// ---------------------------------------------------------------------------
__global__ void __launch_bounds__(256)
main_kernel(const int* __restrict__ bow, const int* __restrict__ bidx,
            const float* __restrict__ seeds, const float* __restrict__ exp_m,
            const float* __restrict__ exp_n, const float* __restrict__ exp_s,
            const float* __restrict__ pi, const float* __restrict__ seed_word,
            const float* __restrict__ exp_n_sum,
            const float* __restrict__ exp_s_sum, const int* __restrict__ Scnt,
            float* __restrict__ temp_m, float* __restrict__ temp_n,
            float* __restrict__ temp_s, float* __restrict__ gsr_sum,
            float* __restrict__ qz, int B, int V, int K) {
    __shared__ float theta_s[MAXB * MAXK];   // theta[b*K+k]
    __shared__ float tm_s[MAXB * MAXK];      // temp_exp_m partial
    __shared__ float bowt_s[MAXB * BOWP];    // bow tile [b][vi], padded
    __shared__ float seeds_s[MAXK], st_s[MAXK], rt_s[MAXK], pi_s[MAXK];
    __shared__ float tn_s[MAXK], ts_s[MAXK], gsum_s[MAXK];
    __shared__ float red_s[256];
    __shared__ float sw_s;

    const int t  = threadIdx.x;
    const int nt = blockDim.x;               // 256
    const int b  = t >> 2;                   // 0..63
    const int kq = t & 3;

    const float mu_sum   = MU * (float)(*Scnt);
    const float beta_sum = BETA * (float)V;

    const int v0 = blockIdx.x * VPB;

    // block init: theta, tm, bow tile, pi
    for (int i = t; i < B * K; i += nt) {
        int bb = i / K, k = i - bb * K;
        theta_s[i] = exp_m[bidx[bb] * K + k] + ETA;
        tm_s[i]    = 0.0f;
    }
    for (int i = t; i < B * VPB; i += nt) {
        int bb = i / VPB, j = i - bb * VPB;
        int v  = v0 + j;
        bowt_s[bb * BOWP + j] = (bb < B && v < V) ? (float)bow[bb * V + v] : 0.0f;
    }
    if (t < K) { pi_s[t] = pi[t]; gsum_s[t] = 0.0f; }
    __syncthreads();

    float qz_local = 0.0f;

    for (int vi = 0; vi < VPB; ++vi) {
        int v = v0 + vi;
        if (v >= V) break;   // uniform within block

        if (t < K) {
            float es = exp_s[v * K + t];
            float en = exp_n[v * K + t];
            seeds_s[t] = seeds[v * K + t];
            st_s[t]    = (MU + es)   / (mu_sum  + exp_s_sum[t]);
            rt_s[t]    = (BETA + en) / (beta_sum + exp_n_sum[t]);
            tn_s[t]    = 0.0f;
            ts_s[t]    = 0.0f;
        }
        if (t == 0) {
            sw_s = seed_word[v];
            if (v + 1 < V) {   // pull next v's rows toward the caches
                __builtin_prefetch(&exp_s[(v + 1) * K], 0, 1);
                __builtin_prefetch(&exp_n[(v + 1) * K], 0, 1);
                __builtin_prefetch(&seeds[(v + 1) * K], 0, 1);
            }
        }
        __syncthreads();

        const float bowf = (b < B) ? bowt_s[b * BOWP + vi] : 0.0f;
        const float sw   = sw_s;
        if (bowf > 0.0f) {
            // pass A: per-(b,v) k-sums (4 lanes per b, k strided by 4)
            float sp = 0.0f, rp = 0.0f;
            for (int k = kq; k < K; k += 4) {
                float th  = theta_s[b * K + k];
                float sd  = seeds_s[k];
                float pik = pi_s[k];
                float gss = sd * th * st_s[k] * pik;
                float gsr = sd * th * rt_s[k] * (1.0f - pik);
                float grr = (1.0f - sd) * th * rt_s[k];
                sp += gss + gsr;
                rp += grr;
            }
            sp += __shfl_xor(sp, 1);  sp += __shfl_xor(sp, 2);
            rp += __shfl_xor(rp, 1);  rp += __shfl_xor(rp, 2);
            const float inv_s = 1.0f / (sp + MINI);
            const float inv_r = 1.0f / (rp + MINI);

            // pass B: normalized gammas + accumulation
            for (int k = kq; k < K; k += 4) {
                float th  = theta_s[b * K + k];
                float sd  = seeds_s[k];
                float pik = pi_s[k];
                float st  = st_s[k];
                float rt  = rt_s[k];
                float gss = sd * th * st * pik * inv_s;
                float gsr = sd * th * rt * (1.0f - pik) * inv_s;
                float grr = (1.0f - sd) * th * rt * inv_r;
                float gam = (sw > 0.0f)
                              ? (pik * gss + (1.0f - pik) * (gsr + grr))
                              : grr;
                tm_s[b * K + k] += gam * bowf;            // exclusive per thread
                atomicAdd(&tn_s[k], (gsr + grr) * bowf);  // reduce over b
                atomicAdd(&ts_s[k], gss * bowf);
                atomicAdd(&gsum_s[k], gsr);
                qz_local += gam * logf(gam + MINI);
            }
        }
        __syncthreads();
        if (t < K) {               // block owns v: plain stores
            temp_n[v * K + t] = tn_s[t];
            temp_s[v * K + t] = ts_s[t];
        }
        // next iteration's (t<K) zeroing is done by the same threads that
        // just stored, and all other writes are fenced by the barriers above.
    }

    __syncthreads();
    for (int i = t; i < B * K; i += nt) atomicAdd(&temp_m[i], tm_s[i]);
    if (t < K) atomicAdd(&gsr_sum[t], gsum_s[t]);

    red_s[t] = qz_local;
    __syncthreads();
    for (int s = nt / 2; s > 0; s >>= 1) {
        if (t < s) red_s[t] += red_s[t + s];
        __syncthreads();
    }
    if (t == 0) atomicAdd(qz, red_s[0]);
}

// ---------------------------------------------------------------------------
// 5) exp_n_new / exp_s_new (EMA with scale)
// ---------------------------------------------------------------------------
__global__ void final_ns_kernel(const float* __restrict__ exp_n,
                                const float* __restrict__ exp_s,
                                const float* __restrict__ temp_n,
                                const float* __restrict__ temp_s,
                                const int* __restrict__ iter_n,
                                const int* __restrict__ batchC,
                                float* __restrict__ out_n,
                                float* __restrict__ out_s, int VK) {
    int i = blockIdx.x * blockDim.x + threadIdx.x;
    if (i >= VK) return;
    float rho   = 1.0f / powf((float)(*iter_n + 5), 0.9f);
    float scale = C_TOTAL / (float)(*batchC);
    out_n[i] = (1.0f - rho) * exp_n[i] + rho * temp_n[i] * scale;
    out_s[i] = (1.0f - rho) * exp_s[i] + rho * temp_s[i] * scale;
}

// ---------------------------------------------------------------------------
// 6) exp_m_batch_new, pi_new, exp_q_z writeout
// ---------------------------------------------------------------------------
__global__ void final_m_pi_kernel(const float* __restrict__ exp_m,
                                  const int* __restrict__ bidx,
                                  const float* __restrict__ temp_m,
                                  const int* __restrict__ iter_n,
                                  const float* __restrict__ exp_s_sum_new,
                                  const float* __restrict__ gsr_sum,
                                  const float* __restrict__ qz,
                                  float* __restrict__ out_m,
                                  float* __restrict__ out_pi,
                                  float* __restrict__ out_qz, int B, int K) {
    int i = blockIdx.x * blockDim.x + threadIdx.x;
    float rho = 1.0f / powf((float)(*iter_n + 5), 0.9f);
    if (i < B * K) {
        int b = i / K, k = i - b * K;
        out_m[i] = (1.0f - rho) * exp_m[bidx[b] * K + k] + rho * temp_m[i];
    }
    if (i < K) {
        float ssn = exp_s_sum_new[i];
        float pn  = ssn / (ssn + gsr_sum[i] + MINI);
        out_pi[i] = (pn > 0.1f) ? pn : PI_INIT;
    }
    if (i == 0) *out_qz = *qz;
}

// ---------------------------------------------------------------------------
// launch
// ---------------------------------------------------------------------------
extern "C" void kernel_launch(void* const* d_in, const int* in_sizes, int n_in,
                              void* d_out, int out_size, void* d_ws,
                              size_t ws_size, hipStream_t stream) {
    const int*   bow    = (const int*)d_in[0];
    const int*   bidx   = (const int*)d_in[1];
    const float* seeds  = (const float*)d_in[2];
    const float* exp_m  = (const float*)d_in[3];
    const float* exp_n  = (const float*)d_in[4];
    const float* exp_s  = (const float*)d_in[5];
    const float* pi     = (const float*)d_in[6];
    const int*   iter_n = (const int*)d_in[7];

    const int B  = in_sizes[1];
    const int K  = in_sizes[6];
    const int V  = in_sizes[2] / K;
    const int VK = V * K;

    // workspace layout (floats)
    float* ws            = (float*)d_ws;
    float* temp_m        = ws;                  // B*K
    float* temp_n        = temp_m + B * K;      // VK
    float* temp_s        = temp_n + VK;         // VK
    float* gsr_sum       = temp_s + VK;         // K
    float* qz            = gsr_sum + K;         // 1
    int*   Scnt          = (int*)(qz + 1);      // 1
    int*   batchC        = Scnt + 1;            // 1
    float* exp_n_sum     = (float*)(batchC + 1);// K
    float* exp_s_sum     = exp_n_sum + K;       // K
    float* exp_s_sum_new = exp_s_sum + K;       // K
    float* seed_word     = exp_s_sum_new + K;   // V

    // output layout (floats, return order)
    float* out_m  = (float*)d_out;      // B*K
    float* out_n  = out_m + B * K;      // VK
    float* out_s  = out_n + VK;         // VK
    float* out_pi = out_s + VK;         // K
    float* out_qz = out_pi + K;         // 1

    // 0) zero accumulators (incl. int slots: bit pattern of 0.0f == 0)
    const int nz = B * K + 2 * VK + 4 * K + 3;
    zero_kernel<<<(nz + 255) / 256, 256, 0, stream>>>(ws, nz);

    // 1) seed_word + S
    seed_word_kernel<<<(V + 255) / 256, 256, 0, stream>>>(seeds, seed_word,
                                                          Scnt, V, K);
    // 2) batch_C
    batchc_kernel<<<256, 256, 0, stream>>>(bow, batchC, B * V);

    // 3) column sums via WMMA ones-trick
    colsum_wmma_kernel<<<64, 128, 0, stream>>>(exp_n, exp_n_sum, V, K);
    colsum_wmma_kernel<<<64, 128, 0, stream>>>(exp_s, exp_s_sum, V, K);

    // 4) fused gamma / normalization / reductions
    main_kernel<<<(V + VPB - 1) / VPB, 256, 0, stream>>>(
        bow, bidx, seeds, exp_m, exp_n, exp_s, pi, seed_word, exp_n_sum,
        exp_s_sum, Scnt, temp_m, temp_n, temp_s, gsr_sum, qz, B, V, K);

    // 5) exp_n_new / exp_s_new
    final_ns_kernel<<<(VK + 255) / 256, 256, 0, stream>>>(
        exp_n, exp_s, temp_n, temp_s, iter_n, batchC, out_n, out_s, VK);

    // 6) exp_s_sum_new = colsum(exp_s_new)  (WMMA again)
    colsum_wmma_kernel<<<64, 128, 0, stream>>>(out_s, exp_s_sum_new, V, K);

    // 7) exp_m_batch_new, pi_new, exp_q_z
    final_m_pi_kernel<<<(B * K + 255) / 256, 256, 0, stream>>>(
        exp_m, bidx, temp_m, iter_n, exp_s_sum_new, gsr_sum, qz, out_m, out_pi,
        out_qz, B, K);
}